// TverskyProjection_80281528697066
// MI455X (gfx1250) — compile-verified
//
#include <hip/hip_runtime.h>
#include <math.h>

typedef __attribute__((ext_vector_type(2))) float v2f;
typedef __attribute__((ext_vector_type(8))) float v8f;

#define IN_F   1024
#define NUM_F  256
#define OUT_F  512
#define ROWS   512   // 2 * 256 flattened batch rows

__device__ __forceinline__ float softplusf(float v) {
    // logaddexp(v, 0) = max(v,0) + log1p(exp(-|v|))  (matches jax.nn.softplus)
    return fmaxf(v, 0.0f) + log1pf(__expf(-fabsf(v)));
}

// ---------------------------------------------------------------------------
// Stage 1: XF[ROWS, NUM_F] = softplus(X[ROWS, IN_F] @ FB[IN_F, NUM_F])
// One wave32 per 16x16 output tile, V_WMMA_F32_16X16X4_F32 over K in steps of 4.
// A layout (16x4 f32, 2 VGPRs): lanes 0-15 -> M=lane, K={0,1}; lanes 16-31 -> M=lane-16, K={2,3}
// B layout (4x16 f32, 2 VGPRs): lanes 0-15 -> N=lane, K={0,1}; lanes 16-31 -> N=lane-16, K={2,3}
// C/D (16x16 f32, 8 VGPRs): vgpr r -> M = r + 8*(lane>=16), N = lane&15
// ---------------------------------------------------------------------------
__global__ void feat_gemm_wmma_kernel(const float* __restrict__ X,
                                      const float* __restrict__ FB,
                                      float* __restrict__ XF) {
    const int wave = threadIdx.x >> 5;
    const int lane = threadIdx.x & 31;
    const int tile = blockIdx.x * 8 + wave;      // 512 tiles total
    const int m0   = (tile >> 4) << 4;           // 32 M-tiles
    const int n0   = (tile & 15) << 4;           // 16 N-tiles
    const int mr   = lane & 15;
    const int hi   = lane >> 4;                  // 0 or 1
    const int klo  = hi * 2;                     // K sub-offset for this half-wave

    const float* arow = X  + (m0 + mr) * IN_F + klo;   // A: row m0+mr, starting K
    const float* bcol = FB + n0 + mr;                  // B: column n0+mr

    v8f acc = {};
    for (int kk = 0; kk < IN_F; kk += 4) {
        v2f a, b;
        a.x = arow[kk + 0];
        a.y = arow[kk + 1];
        b.x = bcol[(kk + klo)     * NUM_F];
        b.y = bcol[(kk + klo + 1) * NUM_F];
        // (neg_a, A, neg_b, B, c_mod, C, reuse_a, reuse_b)
        acc = __builtin_amdgcn_wmma_f32_16x16x4_f32(false, a, false, b,
                                                    (short)0, acc, false, false);
    }

    const int ncol = n0 + mr;
#pragma unroll
    for (int r = 0; r < 8; ++r) {
        const int m = m0 + r + hi * 8;
        XF[m * NUM_F + ncol] = softplusf(acc[r]);
    }
}

// ---------------------------------------------------------------------------
// Stage 2: PF = softplus(prototype_bank), elementwise 512x256
// ---------------------------------------------------------------------------
__global__ void proto_softplus_kernel(const float* __restrict__ PB,
                                      float* __restrict__ PF) {
    const int idx = blockIdx.x * 256 + threadIdx.x;
    PF[idx] = softplusf(PB[idx]);
}

// ---------------------------------------------------------------------------
// Stage 3: pairwise Tversky similarity.
//   d  = xe - pe
//   wx = sigmoid((pe-xe)/0.5) = sigmoid(-2d)
//   common = wx*xe + (1-wx)*pe = pe + sigmoid(-2d)*d
//   x_dist = relu(d), p_dist = relu(-d)
// Block: 256 threads = 256 prototypes; grid (2, 512): j-chunk x row i.
// x row staged in LDS; prototype rows are L2-resident (512 KB total).
// ---------------------------------------------------------------------------
__global__ void tversky_pairwise_kernel(const float* __restrict__ XF,
                                        const float* __restrict__ PF,
                                        const float* __restrict__ alpha_raw,
                                        const float* __restrict__ beta_raw,
                                        const float* __restrict__ gamma_p,
                                        const float* __restrict__ bias,
                                        float* __restrict__ out) {
    __shared__ __align__(16) float xs[NUM_F];
    const int i = blockIdx.y;                       // 0..511 flattened batch row
    const int j = blockIdx.x * 256 + threadIdx.x;   // 0..511 prototype

    xs[threadIdx.x] = XF[i * NUM_F + threadIdx.x];  // NUM_F == blockDim.x == 256
    __syncthreads();

    const float4* pr = (const float4*)(PF + j * NUM_F);
    const float4* xr = (const float4*)xs;

    float accC = 0.0f, accX = 0.0f, accP = 0.0f;
#pragma unroll 4
    for (int k4 = 0; k4 < NUM_F / 4; ++k4) {
        const float4 p  = pr[k4];
        const float4 xv = xr[k4];
        {
            const float d = xv.x - p.x;
            const float s = 1.0f / (1.0f + __expf(2.0f * d));
            accC += p.x + s * d; accX += fmaxf(d, 0.0f); accP += fmaxf(-d, 0.0f);
        }
        {
            const float d = xv.y - p.y;
            const float s = 1.0f / (1.0f + __expf(2.0f * d));
            accC += p.y + s * d; accX += fmaxf(d, 0.0f); accP += fmaxf(-d, 0.0f);
        }
        {
            const float d = xv.z - p.z;
            const float s = 1.0f / (1.0f + __expf(2.0f * d));
            accC += p.z + s * d; accX += fmaxf(d, 0.0f); accP += fmaxf(-d, 0.0f);
        }
        {
            const float d = xv.w - p.w;
            const float s = 1.0f / (1.0f + __expf(2.0f * d));
            accC += p.w + s * d; accX += fmaxf(d, 0.0f); accP += fmaxf(-d, 0.0f);
        }
    }

    const float alpha = softplusf(alpha_raw[0]);
    const float beta  = softplusf(beta_raw[0]);
    const float g     = gamma_p[0];
    const float sim   = g * accC / (accC + alpha * accX + beta * accP + 1e-8f);
    out[i * OUT_F + j] = sim + bias[j];
}

extern "C" void kernel_launch(void* const* d_in, const int* in_sizes, int n_in,
                              void* d_out, int out_size, void* d_ws, size_t ws_size,
                              hipStream_t stream) {
    const float* x         = (const float*)d_in[0];
    const float* fb        = (const float*)d_in[1];
    const float* pb        = (const float*)d_in[2];
    const float* alpha_raw = (const float*)d_in[3];
    const float* beta_raw  = (const float*)d_in[4];
    const float* gamma_p   = (const float*)d_in[5];
    const float* bias      = (const float*)d_in[6];
    float* out = (float*)d_out;

    float* XF = (float*)d_ws;                 // 512*256 floats = 512 KB
    float* PF = XF + (size_t)ROWS * NUM_F;    // 512*256 floats = 512 KB

    // Stage 1: 512 wave-tiles -> 64 blocks x 8 waves (256 threads)
    feat_gemm_wmma_kernel<<<64, 256, 0, stream>>>(x, fb, XF);

    // Stage 2: 512*256 elements / 256 threads
    proto_softplus_kernel<<<(OUT_F * NUM_F) / 256, 256, 0, stream>>>(pb, PF);

    // Stage 3: grid (j-chunks=2, rows=512), 256 threads
    dim3 g3(2, ROWS);
    tversky_pairwise_kernel<<<g3, 256, 0, stream>>>(XF, PF, alpha_raw, beta_raw,
                                                    gamma_p, bias, out);
}